// ContextAttentionModule_26938034881104
// MI455X (gfx1250) — compile-verified
//
#include <hip/hip_runtime.h>
#include <math.h>

#define BATCH  8
#define CH     256
#define HW     16384      // 128*128
#define KSEL   64
#define SMOOTH 1e-6f

typedef __attribute__((ext_vector_type(2))) float v2f;
typedef __attribute__((ext_vector_type(8))) float v8f;

// sigmoid via v_exp_f32 + v_rcp_f32 (avoids the IEEE div_scale/div_fmas
// expansion: ~10 VALU ops saved per element in the 33.5M-element pass 1)
__device__ __forceinline__ float fast_sigmoid(float v) {
    return __builtin_amdgcn_rcpf(1.0f + __expf(-v));
}

// ---------------------------------------------------------------------------
// Pass 1: score[b,c] = mean_{hw} ( -s * log(s + SMOOTH) ),  s = sigmoid(x)
// One block per (b,c) channel image (16384 floats = 4096 float4).
// ---------------------------------------------------------------------------
__global__ void score_kernel(const float* __restrict__ x,
                             float* __restrict__ score) {
    const unsigned bc  = blockIdx.x;                  // 0 .. B*C-1
    const unsigned tid = threadIdx.x;                 // 256 threads
    const float4* xp = (const float4*)(x + (size_t)bc * HW);

    float acc = 0.0f;
#pragma unroll
    for (int i = 0; i < 16; ++i) {                    // 16 * 256 = 4096 float4
        float4 v = xp[i * 256 + tid];
        float e[4] = {v.x, v.y, v.z, v.w};
#pragma unroll
        for (int j = 0; j < 4; ++j) {
            float s = fast_sigmoid(e[j]);
            acc += -s * __logf(s + SMOOTH);
        }
    }

    __shared__ float red[256];
    red[tid] = acc;
    __syncthreads();
#pragma unroll
    for (int s = 128; s > 0; s >>= 1) {
        if (tid < (unsigned)s) red[tid] += red[tid + s];
        __syncthreads();
    }
    if (tid == 0) score[bc] = red[0] * (1.0f / (float)HW);
}

// ---------------------------------------------------------------------------
// Pass 2: per batch, indices of the 64 smallest scores, in ascending-score
// order with stable (lower-index-first) tie-breaking == jax.lax.top_k(-score).
// One block of 256 threads per batch; O(256^2) rank computation.
// ---------------------------------------------------------------------------
__global__ void topk_kernel(const float* __restrict__ score,
                            int* __restrict__ sel) {
    const int b = blockIdx.x;
    const int c = threadIdx.x;
    __shared__ float sc[CH];
    sc[c] = score[b * CH + c];
    __syncthreads();

    const float my = sc[c];
    int rank = 0;
    for (int j = 0; j < CH; ++j) {
        float o = sc[j];
        rank += (o < my) || (o == my && j < c);
    }
    if (rank < KSEL) sel[b * KSEL + rank] = c;
}

// ---------------------------------------------------------------------------
// Pass 3: attn[b,hw] = sigmoid( sum_k x[b, sel[b,k], hw] * w[k] + bias )
// via V_WMMA_F32_16X16X4_F32.  Each wave32 computes 16 pixels:
//   A (16x4)  = w[k-chunk] broadcast down all 16 rows
//   B (4x16)  = selected-channel data for 16 contiguous pixels
//   K = 64 accumulated over 16 WMMA steps of K=4 (exact fp32).
// VGPR layouts per ISA 7.12.2:
//   A: lane L, vgpr v  -> K = 2*(L>>4) + v            (M = L&15, rows equal)
//   B: lane L, vgpr v  -> K = 2*(L>>4) + v, N = L&15
//   D: vgpr0, lanes 0-15 -> row M=0, N = lane  (all rows identical here)
// Block = 256 threads = 8 waves = 128 pixels; blocks never straddle batches.
// ---------------------------------------------------------------------------
__global__ void attn_kernel(const float* __restrict__ x,
                            const float* __restrict__ w,
                            const float* __restrict__ bias,
                            const int* __restrict__ sel,
                            float* __restrict__ attn) {
    __shared__ float wsh[KSEL];
    __shared__ int   csh[KSEL];

    const int tid = threadIdx.x;
    const unsigned gw0 = blockIdx.x * 8;              // first wave-group id
    const int b = (int)((gw0 * 16) / HW);             // 128 | 16384 -> no straddle

    if (tid < KSEL) {
        wsh[tid] = w[tid];
        csh[tid] = sel[b * KSEL + tid];
    }
    __syncthreads();                                  // EXEC back to all-ones

    const int wave   = tid >> 5;
    const int lane   = tid & 31;
    const unsigned gp = gw0 + wave;                   // 16-pixel group id
    const int hwbase = (int)(gp % (HW / 16)) * 16;
    const int n      = lane & 15;                     // pixel within group
    const int khalf  = lane >> 4;                     // 0 or 1

    const float* xb = x + (size_t)b * CH * HW;

    v8f acc = 0.0f;
#pragma unroll
    for (int t = 0; t < 16; ++t) {
        const int kk = 4 * t + 2 * khalf;             // this lane's K pair
        v2f a, bm;
        a.x  = wsh[kk];
        a.y  = wsh[kk + 1];
        bm.x = xb[(size_t)csh[kk]     * HW + hwbase + n];
        bm.y = xb[(size_t)csh[kk + 1] * HW + hwbase + n];
        acc = __builtin_amdgcn_wmma_f32_16x16x4_f32(
                  false, a, false, bm, (short)0, acc, false, false);
    }

    if (lane < 16) {
        float v = acc[0] + bias[0];                   // D row 0, col n
        attn[(size_t)b * HW + hwbase + n] = fast_sigmoid(v);
    }
}

// ---------------------------------------------------------------------------
// Pass 4: out[b,c,hw] = x[b,c,hw] * attn[b,hw]   (float4 streaming)
// attn is 512 KB -> L2-resident across all 256 channel re-reads.
// ---------------------------------------------------------------------------
__global__ void mul_kernel(const float* __restrict__ x,
                           const float* __restrict__ attn,
                           float* __restrict__ out) {
    const unsigned i    = blockIdx.x * blockDim.x + threadIdx.x; // float4 id
    const unsigned flat = i * 4;
    const unsigned pix  = flat & (HW - 1);
    const unsigned bidx = flat >> 22;                 // / (CH*HW) = / 2^22

    const float4 xv = ((const float4*)x)[i];
    const float4 av = *(const float4*)(attn + (size_t)bidx * HW + pix);
    float4 o;
    o.x = xv.x * av.x;
    o.y = xv.y * av.y;
    o.z = xv.z * av.z;
    o.w = xv.w * av.w;
    ((float4*)out)[i] = o;
}

// ---------------------------------------------------------------------------
extern "C" void kernel_launch(void* const* d_in, const int* in_sizes, int n_in,
                              void* d_out, int out_size, void* d_ws, size_t ws_size,
                              hipStream_t stream) {
    const float* x    = (const float*)d_in[0];   // [8,256,128,128]
    const float* w    = (const float*)d_in[1];   // [64]
    const float* bias = (const float*)d_in[2];   // [1]
    float* out = (float*)d_out;

    float* score = (float*)d_ws;                        // 2048 f32  @ 0
    int*   sel   = (int*)((char*)d_ws + 8192);          // 512  i32  @ 8 KB
    float* attn  = (float*)((char*)d_ws + 16384);       // 131072 f32 @ 16 KB

    score_kernel<<<BATCH * CH, 256, 0, stream>>>(x, score);
    topk_kernel<<<BATCH, CH, 0, stream>>>(score, sel);
    attn_kernel<<<(BATCH * HW) / 128, 256, 0, stream>>>(x, w, bias, sel, attn);
    mul_kernel<<<(BATCH * CH * HW) / (4 * 256), 256, 0, stream>>>(x, attn, out);
}